// CAModel_29549374996729
// MI455X (gfx1250) — compile-verified
//
#include <hip/hip_runtime.h>
#include <hip/hip_bf16.h>

// Problem constants (fixed by the reference)
#define B_  2
#define S_  160
#define H_  768
#define C_  5
#define NN_ (C_ * H_)   // 3840
#define M_  (B_ * S_)   // 320

typedef __attribute__((ext_vector_type(2))) float v2f;
typedef __attribute__((ext_vector_type(8))) float v8f;

// ---------------------------------------------------------------------------
// Kernel 1: fp32 projection GEMM via V_WMMA_F32_16X16X4_F32.
//   dst[m, n] = seq[m, :] . W[:, n] + bias[n]
//   M=320, N=3840, K=768. One wave per 16x32 output strip (two 16x16 tiles,
//   A fragment reused across both -> 4 WMMAs per unrolled loop body).
//   grid = (3840/32/8, 20, 2): z selects (w_prd,b_prd,hP) vs (w_arg,b_arg,hA).
// ---------------------------------------------------------------------------
__global__ __launch_bounds__(256)
void proj_gemm_wmma(const float* __restrict__ seq,
                    const float* __restrict__ wPrd, const float* __restrict__ bPrd,
                    const float* __restrict__ wArg, const float* __restrict__ bArg,
                    float* __restrict__ hP, float* __restrict__ hA) {
  const float* W    = (blockIdx.z == 0) ? wPrd : wArg;
  const float* bias = (blockIdx.z == 0) ? bPrd : bArg;
  float*       dst  = (blockIdx.z == 0) ? hP   : hA;

  const int wave  = threadIdx.x >> 5;
  const int lane  = threadIdx.x & 31;
  const int nBase = (blockIdx.x * 8 + wave) * 32;   // 2 adjacent 16-wide tiles
  const int mBase = blockIdx.y * 16;

  const int mi  = lane & 15;   // M (for A) / N (for B) within tile
  const int sel = lane >> 4;   // selects K-pair {0,1} vs {2,3}

  // A fragment source: A[mBase+mi][k + 2*sel + {0,1}]
  const float* aPtr = seq + (size_t)(mBase + mi) * H_ + 2 * sel;
  // B fragment source: W[k + 2*sel + {0,1}][nBase + mi (+16)]
  const float* bPtr = W + (size_t)(2 * sel) * NN_ + nBase + mi;

  v8f acc0 = {};
  v8f acc1 = {};
#pragma unroll 2
  for (int k = 0; k < H_; k += 4) {
    v2f a, b0, b1;
    a.x  = aPtr[k];
    a.y  = aPtr[k + 1];
    b0.x = bPtr[(size_t)k * NN_];
    b0.y = bPtr[(size_t)(k + 1) * NN_];
    b1.x = bPtr[(size_t)k * NN_ + 16];
    b1.y = bPtr[(size_t)(k + 1) * NN_ + 16];
    acc0 = __builtin_amdgcn_wmma_f32_16x16x4_f32(
        false, a, false, b0, (short)0, acc0, false, false);
    acc1 = __builtin_amdgcn_wmma_f32_16x16x4_f32(
        false, a, false, b1, (short)0, acc1, false, false);
  }

  const int   col  = nBase + mi;
  const float bc0  = bias[col];
  const float bc1  = bias[col + 16];
#pragma unroll
  for (int r = 0; r < 8; ++r) {
    const int row = mBase + r + 8 * sel;           // C/D layout: VGPR r -> M = r + 8*lane[4]
    dst[(size_t)row * NN_ + col]      = acc0[r] + bc0;
    dst[(size_t)row * NN_ + col + 16] = acc1[r] + bc1;
  }
}

// ---------------------------------------------------------------------------
// Kernel 2: fused tanh cube + rank-1 reduction + mask penalty.
//   out[b,p,c,a] = sum_h w_out[h] * tanh(hP[b,p,c,h] + hA[b,a,c,h])
//                  + (mask ? 0 : -1024)
//   grid = (S/16, S/16, B*C), block = 256 = 16x16 (p x a) tile.
// ---------------------------------------------------------------------------
__device__ __forceinline__ float fast_tanh(float x) {
#if __has_builtin(__builtin_amdgcn_tanhf)
  return __builtin_amdgcn_tanhf(x);                 // single TRANS32 v_tanh_f32
#elif __has_builtin(__builtin_amdgcn_tanh_f32)
  return __builtin_amdgcn_tanh_f32(x);
#else
  x = fminf(fmaxf(x, -15.f), 15.f);                 // tanh saturates; avoid inf/inf
  const float e = __expf(2.f * x);
  return (e - 1.f) / (e + 1.f);
#endif
}

#define KC 64
__global__ __launch_bounds__(256)
void pas_score(const float* __restrict__ hP, const float* __restrict__ hA,
               const float* __restrict__ wOut, const unsigned char* __restrict__ mask,
               float* __restrict__ outbuf) {
  __shared__ float sP[16][KC + 1];   // +1 pad: distinct banks across rows
  __shared__ float sA[16][KC + 1];
  __shared__ float sW[KC];

  const int bc = blockIdx.z;
  const int b  = bc / C_;
  const int c  = bc % C_;
  const int p0 = blockIdx.y * 16;
  const int a0 = blockIdx.x * 16;

  const int tx = threadIdx.x & 15;   // a within tile
  const int ty = threadIdx.x >> 4;   // p within tile

  const float* rowP = hP + ((size_t)(b * S_ + p0) * C_ + c) * H_;  // +r*C_*H_ per p
  const float* rowA = hA + ((size_t)(b * S_ + a0) * C_ + c) * H_;

  float acc = 0.f;
  for (int kb = 0; kb < H_; kb += KC) {
    __syncthreads();
    for (int i = threadIdx.x; i < 16 * KC; i += 256) {
      const int r = i >> 6;          // row in tile
      const int k = i & (KC - 1);
      sP[r][k] = rowP[(size_t)r * (C_ * H_) + kb + k];
      sA[r][k] = rowA[(size_t)r * (C_ * H_) + kb + k];
    }
    if (threadIdx.x < KC) sW[threadIdx.x] = wOut[kb + threadIdx.x];
    // Prefetch next chunk (one global_prefetch_b8 per 64B line, 4 lines/row)
    if (kb + KC < H_ && threadIdx.x < 64) {
      const int r = threadIdx.x >> 2;
      const int l = (threadIdx.x & 3) * 16;
      __builtin_prefetch(&rowP[(size_t)r * (C_ * H_) + kb + KC + l], 0, 0);
      __builtin_prefetch(&rowA[(size_t)r * (C_ * H_) + kb + KC + l], 0, 0);
    }
    __syncthreads();
#pragma unroll 8
    for (int k = 0; k < KC; ++k)
      acc = fmaf(sW[k], fast_tanh(sP[ty][k] + sA[tx][k]), acc);
  }

  const int p = p0 + ty;
  const int a = a0 + tx;
  const size_t oidx = ((size_t)(b * S_ + p) * C_ + c) * S_ + a;  // [B,Sp,C,Sa] flat
  const float pen = mask[oidx] ? 0.f : -1024.f;
  outbuf[1 + oidx] = acc + pen;      // outbuf[0] holds the loss
}

// ---------------------------------------------------------------------------
// Kernel 3: per-row (b,p,c) log-softmax CE partials over the Sa axis.
//   grid = 1600 rows, block = 256 (160 active lanes).
// ---------------------------------------------------------------------------
__global__ __launch_bounds__(256)
void row_loss(const float* __restrict__ outbuf, const int* __restrict__ target,
              float* __restrict__ rowNum, float* __restrict__ rowDen) {
  __shared__ float red[256];
  const int r   = blockIdx.x;
  const int tid = threadIdx.x;
  const float* row = outbuf + 1 + (size_t)r * S_;

  const float v = (tid < S_) ? row[tid] : -3.0e38f;
  red[tid] = v; __syncthreads();
  for (int s = 128; s > 0; s >>= 1) {
    if (tid < s) red[tid] = fmaxf(red[tid], red[tid + s]);
    __syncthreads();
  }
  const float m = red[0]; __syncthreads();

  const float e = (tid < S_) ? __expf(v - m) : 0.f;
  red[tid] = e; __syncthreads();
  for (int s = 128; s > 0; s >>= 1) {
    if (tid < s) red[tid] += red[tid + s];
    __syncthreads();
  }
  const float logZ = m + __logf(red[0]); __syncthreads();

  const float t = (tid < S_) ? (float)target[(size_t)r * S_ + tid] : 0.f;
  const float numv = (tid < S_) ? t * (logZ - v) : 0.f;   // -log_softmax * t

  red[tid] = numv; __syncthreads();
  for (int s = 128; s > 0; s >>= 1) {
    if (tid < s) red[tid] += red[tid + s];
    __syncthreads();
  }
  if (tid == 0) rowNum[r] = red[0];
  __syncthreads();

  red[tid] = t; __syncthreads();
  for (int s = 128; s > 0; s >>= 1) {
    if (tid < s) red[tid] += red[tid + s];
    __syncthreads();
  }
  if (tid == 0) rowDen[r] = red[0];
}

// ---------------------------------------------------------------------------
// Kernel 4: final reduction of 1600 row partials -> loss at d_out[0].
// ---------------------------------------------------------------------------
__global__ __launch_bounds__(256)
void finalize_loss(const float* __restrict__ rowNum, const float* __restrict__ rowDen,
                   float* __restrict__ outbuf) {
  __shared__ float rn[256], rd[256];
  const int tid = threadIdx.x;
  float n = 0.f, d = 0.f;
  for (int i = tid; i < M_ * C_; i += 256) { n += rowNum[i]; d += rowDen[i]; }
  rn[tid] = n; rd[tid] = d; __syncthreads();
  for (int s = 128; s > 0; s >>= 1) {
    if (tid < s) { rn[tid] += rn[tid + s]; rd[tid] += rd[tid + s]; }
    __syncthreads();
  }
  if (tid == 0) outbuf[0] = rn[0] / (rd[0] + 1e-6f);
}

// ---------------------------------------------------------------------------
extern "C" void kernel_launch(void* const* d_in, const int* in_sizes, int n_in,
                              void* d_out, int out_size, void* d_ws, size_t ws_size,
                              hipStream_t stream) {
  const float* seq   = (const float*)d_in[0];
  const float* w_prd = (const float*)d_in[1];
  const float* b_prd = (const float*)d_in[2];
  const float* w_arg = (const float*)d_in[3];
  const float* b_arg = (const float*)d_in[4];
  const float* w_out = (const float*)d_in[5];
  const unsigned char* mask = (const unsigned char*)d_in[6];  // jax bool = 1 byte
  const int* target  = (const int*)d_in[7];

  float* out = (float*)d_out;          // [0] = loss, [1..] = output [B,Sp,C,Sa]
  float* ws  = (float*)d_ws;
  float* hP     = ws;                              // 320*3840 = 1,228,800 f
  float* hA     = ws + (size_t)M_ * NN_;           // 1,228,800 f
  float* rowNum = ws + 2 * (size_t)M_ * NN_;       // 1600 f
  float* rowDen = rowNum + M_ * C_;                // 1600 f

  // 1) Projections (fp32 WMMA): 120 N-strips of 32 grouped 8/block, 20 M-tiles, 2 mats
  dim3 g1(NN_ / 32 / 8, M_ / 16, 2);
  proj_gemm_wmma<<<g1, 256, 0, stream>>>(seq, w_prd, b_prd, w_arg, b_arg, hP, hA);

  // 2) Fused tanh cube + contraction + mask
  dim3 g2(S_ / 16, S_ / 16, B_ * C_);
  pas_score<<<g2, 256, 0, stream>>>(hP, hA, w_out, mask, out);

  // 3) Per-row softmax CE partials
  row_loss<<<M_ * C_, 256, 0, stream>>>(out, target, rowNum, rowDen);

  // 4) Scalar loss
  finalize_loss<<<1, 256, 0, stream>>>(rowNum, rowDen, out);
}